// NeighborList_64845416235103
// MI455X (gfx1250) — compile-verified
//
#include <hip/hip_runtime.h>

typedef float v2f __attribute__((ext_vector_type(2)));
typedef float v8f __attribute__((ext_vector_type(8)));

#define NATOMS 20000
#define NCELLS 4096
#define KCELL  32     // atoms kept per cell
#define KNBR   26     // neighbor cells per cell
#define KOUT   16     // final neighbors per atom

// ----------------- wave32 helpers -----------------
__device__ __forceinline__ unsigned ballot32(bool p) {
  return __builtin_amdgcn_ballot_w32(p);
}

// min by value, ties -> larger index (the entry we'd rather evict). Branchless.
__device__ __forceinline__ void wave_min_entry(float& v, int& c) {
#pragma unroll
  for (int m = 16; m; m >>= 1) {
    float ov = __shfl_xor(v, m, 32);
    int   oc = __shfl_xor(c, m, 32);
    bool take = (ov < v) | ((ov == v) & (oc > c));
    v = take ? ov : v;
    c = take ? oc : c;
  }
}

// max by value, ties -> smaller index (top_k stable order), carries payload. Branchless.
__device__ __forceinline__ void wave_max_entry(float& v, int& c, int& a) {
#pragma unroll
  for (int m = 16; m; m >>= 1) {
    float ov = __shfl_xor(v, m, 32);
    int   oc = __shfl_xor(c, m, 32);
    int   oa = __shfl_xor(a, m, 32);
    bool take = (ov > v) | ((ov == v) & (oc < c));
    v = take ? ov : v;
    c = take ? oc : c;
    a = take ? oa : a;
  }
}

// Distributed top-k insert: lane t holds list entry t in (lv,lc).
// (mv,mc) is the wave-uniform current worst entry. One candidate per lane.
__device__ __forceinline__ void topk_insert(float v, int c, bool valid,
                                            float& lv, int& lc,
                                            float& mv, int& mc) {
  bool pred = valid & ((v > mv) | ((v == mv) & (c < mc)));   // non-short-circuit
  unsigned mask = ballot32(pred);
  while (mask) {
    int s = __ffs(mask) - 1;
    mask &= mask - 1;
    float bv = __shfl(v, s, 32);
    int   bc = __shfl(c, s, 32);
    bool ok = (bv > mv) | ((bv == mv) & (bc < mc));           // re-check (mv moved)
    if (ok) {
      bool own = (lv == mv) & (lc == mc);                     // unique owner lane
      lv = own ? bv : lv;
      lc = own ? bc : lc;
      mv = lv; mc = lc;
      wave_min_entry(mv, mc);
    }
  }
}

// Payload variant (stage C carries the gathered atom index).
__device__ __forceinline__ void topk_insert3(float v, int c, int a,
                                             float& lv, int& lc, int& la,
                                             float& mv, int& mc) {
  bool pred = (v > mv) | ((v == mv) & (c < mc));
  unsigned mask = ballot32(pred);
  while (mask) {
    int s = __ffs(mask) - 1;
    mask &= mask - 1;
    float bv = __shfl(v, s, 32);
    int   bc = __shfl(c, s, 32);
    int   ba = __shfl(a, s, 32);
    bool ok = (bv > mv) | ((bv == mv) & (bc < mc));
    if (ok) {
      bool own = (lv == mv) & (lc == mc);
      lv = own ? bv : lv;
      lc = own ? bc : lc;
      la = own ? ba : la;
      mv = lv; mc = lc;
      wave_min_entry(mv, mc);
    }
  }
}

// ----------------- prep kernels -----------------
__global__ void nl_init_kernel(float* wsmin) {
  if (threadIdx.x == 0 && blockIdx.x == 0) ((int*)wsmin)[0] = 0x7f800000; // +inf
}

__global__ void nl_min_kernel(const float* __restrict__ coords, float* wsmin, int n) {
  int t = blockIdx.x * blockDim.x + threadIdx.x;
  float v = __builtin_inff();
  for (int i = t; i < n; i += gridDim.x * blockDim.x) v = fminf(v, coords[i]);
#pragma unroll
  for (int m = 16; m; m >>= 1) v = fminf(v, __shfl_xor(v, m, 32));
  __shared__ float red[8];
  int lane = threadIdx.x & 31, w = threadIdx.x >> 5;
  if (lane == 0) red[w] = v;
  __syncthreads();
  if (threadIdx.x == 0) {
    float b = red[0];
    for (int i = 1; i < (int)(blockDim.x >> 5); ++i) b = fminf(b, red[i]);
    // coords > 0 -> IEEE bit order == float order
    atomicMin((int*)wsmin, __float_as_int(b));
  }
}

// cells[f] = (r[j], r[i], r[k]) with f = k*256 + j*16 + i (meshgrid/transpose order)
// Also emits the -2-scaled copies used as WMMA B operands.
__global__ void nl_cells_kernel(const float* __restrict__ wsmin,
                                float* cx, float* cy, float* cz, float* cn,
                                float* cxs, float* cys, float* czs) {
  int f = blockIdx.x * blockDim.x + threadIdx.x;
  if (f >= NCELLS) return;
  float start = floorf(wsmin[0]);
  int i = f & 15, j = (f >> 4) & 15, k = f >> 8;
  float x = start + (float)j, y = start + (float)i, z = start + (float)k;
  cx[f] = x; cy[f] = y; cz[f] = z;
  cn[f] = x * x + y * y + z * z;
  cxs[f] = -2.0f * x; cys[f] = -2.0f * y; czs[f] = -2.0f * z;
}

// SoA atom coords (+ -2-scaled copies) + norms + nearest-cell id (argmin over cells)
__global__ void nl_atoms_kernel(const float* __restrict__ coords,
                                const float* __restrict__ wsmin,
                                float* ax, float* ay, float* az, float* an,
                                float* axs, float* ays, float* azs,
                                int* cid) {
  int t = blockIdx.x * blockDim.x + threadIdx.x;
  if (t >= NATOMS) return;
  float x = coords[3 * t], y = coords[3 * t + 1], z = coords[3 * t + 2];
  ax[t] = x; ay[t] = y; az[t] = z;
  an[t] = x * x + y * y + z * z;
  axs[t] = -2.0f * x; ays[t] = -2.0f * y; azs[t] = -2.0f * z;
  float start = floorf(wsmin[0]);
  int jx = min(15, max(0, (int)floorf(x - start + 0.5f)));
  int iy = min(15, max(0, (int)floorf(y - start + 0.5f)));
  int kz = min(15, max(0, (int)floorf(z - start + 0.5f)));
  cid[t] = kz * 256 + jx * 16 + iy;
}

// ----------------- fused WMMA pdist2 + per-row top-K -----------------
// Block: 512 threads = 16 waves, owns 16 rows (cells). Each iteration:
//   wave w WMMAs the 16x16 tile at columns colbase + w*16 into LDS,
//   then wave w scans its own row (256 cols) maintaining a register top-K.
// d2[m][n] = ||c_m||^2 - 2 c_m.b_n + ||b_n||^2 via A=(cx,cy,cz,||c||^2),
// B=(-2bx,-2by,-2bz,1) (pre-scaled), C preloaded with ||b_n||^2.
template <int KKEEP>
__global__ void __launch_bounds__(512)
nl_topk_rows_kernel(const float* __restrict__ cx, const float* __restrict__ cy,
                    const float* __restrict__ cz, const float* __restrict__ cn,
                    const float* __restrict__ bxs, const float* __restrict__ bys,
                    const float* __restrict__ bzs, const float* __restrict__ bn,
                    int ncols, int* __restrict__ out) {
  __shared__ float tile[16 * 256];
  const int lane = threadIdx.x & 31;
  const int w = threadIdx.x >> 5;      // wave 0..15 == block row
  const int rowbase = blockIdx.x * 16;
  const int n = lane & 15;
  const bool hi = lane >= 16;

  // A operand (16x4 f32): lanes 0-15 hold K=0,1; lanes 16-31 hold K=2,3.
  // Per-lane pointer select -> one unconditional load each, no EXEC branching.
  const float* pa0 = hi ? cz : cx;
  const float* pa1 = hi ? cn : cy;
  v2f A;
  {
    int m = rowbase + n;
    A[0] = pa0[m];
    A[1] = pa1[m];
  }
  const float* pb0 = hi ? bzs : bxs;   // B K-slot 0/2
  const float* pb1 = bys;              // B K-slot 1 (hi lanes override with 1.0)

  // distributed top-K state: lane t holds entry t; lanes >= KKEEP frozen at +inf
  float lv = (lane < KKEEP) ? -__builtin_inff() : __builtin_inff();
  int   lc = 0x7fffffff - lane;
  float mv = lv; int mc = lc;
  wave_min_entry(mv, mc);

  const int niter = (ncols + 255) >> 8;
  for (int it = 0; it < niter; ++it) {
    const int colbase = it << 8;
    {
      int colc = min(colbase + w * 16 + n, ncols - 1);  // clamp keeps EXEC all-ones
      float t0 = pb0[colc];
      float t1 = pb1[colc];
      float t2 = bn[colc];
      v2f B;
      B[0] = t0;
      B[1] = hi ? 1.0f : t1;
      v8f acc = {t2, t2, t2, t2, t2, t2, t2, t2};
      v8f d = __builtin_amdgcn_wmma_f32_16x16x4_f32(false, A, false, B,
                                                    (short)0, acc, false, false);
      const int rb = hi ? 8 : 0;   // D: VGPR v -> row v (lanes 0-15) / v+8 (16-31)
#pragma unroll
      for (int r = 0; r < 8; ++r)
        tile[(rb + r) * 256 + w * 16 + n] = d[r];
    }
    __syncthreads();
#pragma unroll
    for (int t = 0; t < 8; ++t) {
      int c = colbase + t * 32 + lane;
      float v = tile[w * 256 + t * 32 + lane];
      topk_insert(v, c, c < ncols, lv, lc, mv, mc);
    }
    if (it + 1 < niter) __builtin_prefetch(&bxs[(it + 1) << 8], 0, 0);
    __syncthreads();
  }

  // emit KKEEP indices in top_k order (value desc, index asc)
  float fv = (lane < KKEEP) ? lv : -__builtin_inff();
  int fc = lc, fa = 0;
  const int row = rowbase + w;
  for (int t = 0; t < KKEEP; ++t) {
    float wvv = fv; int wcc = fc; int waa = fa;
    wave_max_entry(wvv, wcc, waa);
    if (lane == 0) out[row * KKEEP + t] = wcc;
    bool win = (fv == wvv) & (fc == wcc);
    fv = win ? -__builtin_inff() : fv;   // retire winner, branchless
  }
}

// ----------------- final per-atom top-16 over 26x32 candidates -----------------
__global__ void __launch_bounds__(256)
nl_final_kernel(const float* __restrict__ ax, const float* __restrict__ ay,
                const float* __restrict__ az, const float* __restrict__ an,
                const int* __restrict__ cid, const int* __restrict__ nbrc,
                const int* __restrict__ aic, int* __restrict__ out) {
  const int lane = threadIdx.x & 31;
  const int w = threadIdx.x >> 5;
  const int atom = blockIdx.x * 8 + w;   // grid = 2500*8 == NATOMS exactly
  const int cell = cid[atom];
  const float x = ax[atom], y = ay[atom], z = az[atom], nx = an[atom];

  float lv = (lane < KOUT) ? -__builtin_inff() : __builtin_inff();
  int lc = 0x7fffffff - lane;
  int la = 0;
  float mv = lv; int mc = lc;
  wave_min_entry(mv, mc);

  __builtin_prefetch(&nbrc[cell * KNBR], 0, 0);
  for (int j = 0; j < KNBR; ++j) {
    int nc = nbrc[cell * KNBR + j];
    int a  = aic[nc * KCELL + lane];   // lane == slot in (26,32) flattening
    float v = nx + an[a] - 2.0f * (x * ax[a] + y * ay[a] + z * az[a]);
    int c = j * 32 + lane;             // flattened candidate position (tie-break key)
    topk_insert3(v, c, a, lv, lc, la, mv, mc);
  }

  float fv = (lane < KOUT) ? lv : -__builtin_inff();
  int fc = lc, fa = la;
  for (int t = 0; t < KOUT; ++t) {
    float wvv = fv; int wcc = fc; int waa = fa;
    wave_max_entry(wvv, wcc, waa);
    if (lane == 0) out[atom * KOUT + t] = waa;   // atom index payload
    bool win = (fv == wvv) & (fc == wcc);
    fv = win ? -__builtin_inff() : fv;
  }
}

// ----------------- launcher -----------------
extern "C" void kernel_launch(void* const* d_in, const int* in_sizes, int n_in,
                              void* d_out, int out_size, void* d_ws, size_t ws_size,
                              hipStream_t stream) {
  const float* coords = (const float*)d_in[0];
  float* ws = (float*)d_ws;

  // workspace layout (floats/ints, ~1.7 MB total — fully L2-resident)
  float* wmin = ws;                       // [1]
  float* cx  = ws + 64;                   // [4096] x4 cell SoA + norm
  float* cy  = cx + NCELLS;
  float* cz  = cy + NCELLS;
  float* cn  = cz + NCELLS;
  float* cxs = cn + NCELLS;               // [4096] x3 cell SoA, -2-scaled
  float* cys = cxs + NCELLS;
  float* czs = cys + NCELLS;
  float* ax  = czs + NCELLS;              // [20000] x4 atom SoA + norm
  float* ay  = ax + NATOMS;
  float* az  = ay + NATOMS;
  float* an  = az + NATOMS;
  float* axs = an + NATOMS;               // [20000] x3 atom SoA, -2-scaled
  float* ays = axs + NATOMS;
  float* azs = ays + NATOMS;
  int* cid  = (int*)(azs + NATOMS);       // [20000] cells_for_atoms
  int* nbrc = cid + NATOMS;               // [4096*26] neighbor_cells
  int* aic  = nbrc + NCELLS * KNBR;       // [4096*32] atoms_in_cells
  int* out  = (int*)d_out;                // [20000*16] neighbor_list (int32)

  nl_init_kernel<<<1, 1, 0, stream>>>(wmin);
  nl_min_kernel<<<64, 256, 0, stream>>>(coords, wmin, NATOMS * 3);
  nl_cells_kernel<<<NCELLS / 256, 256, 0, stream>>>(wmin, cx, cy, cz, cn,
                                                    cxs, cys, czs);
  nl_atoms_kernel<<<(NATOMS + 255) / 256, 256, 0, stream>>>(
      coords, wmin, ax, ay, az, an, axs, ays, azs, cid);
  // neighbor_cells: per-cell top-26 of d2_cc (largest, faithful to reference)
  nl_topk_rows_kernel<KNBR><<<NCELLS / 16, 512, 0, stream>>>(
      cx, cy, cz, cn, cxs, cys, czs, cn, NCELLS, nbrc);
  // atoms_in_cells: per-cell top-32 of d2_ca (largest, faithful to reference)
  nl_topk_rows_kernel<KCELL><<<NCELLS / 16, 512, 0, stream>>>(
      cx, cy, cz, cn, axs, ays, azs, an, NATOMS, aic);
  nl_final_kernel<<<NATOMS / 8, 256, 0, stream>>>(ax, ay, az, an, cid, nbrc, aic, out);

  (void)in_sizes; (void)n_in; (void)out_size; (void)ws_size;
}